// nu_grid_sampler_37890201485784
// MI455X (gfx1250) — compile-verified
//
#include <hip/hip_runtime.h>
#include <math.h>
#include <stdint.h>

// Problem constants (match reference: B,C,NX,NY = 8,16,512,512; N=16384; N_RES=90; NH=5)
#define NU_B    8
#define NU_C    16
#define NU_NX   512
#define NU_NY   512
#define NU_N    16384
#define NU_NRES 90
#define NU_NH   5
#define NU_BIN  (NU_NRES / NU_NH)   // 18
#define NU_TPB  256

// ---- CDNA5 async global->LDS path (guarded: falls back to plain LDS stores) ----
#if defined(__gfx1250__) && __has_builtin(__builtin_amdgcn_global_load_async_to_lds_b64)
#define NU_ASYNC64 1
#else
#define NU_ASYNC64 0
#endif
#if defined(__gfx1250__) && __has_builtin(__builtin_amdgcn_global_load_async_to_lds_b32)
#define NU_ASYNC32 1
#else
#define NU_ASYNC32 0
#endif

#if NU_ASYNC64 || NU_ASYNC32
// Builtin parameter types per hipcc diagnostic:
//   b64: (int __vector(2) * addrspace(1), int __vector(2) * addrspace(3), Ii, Ii)
//   b32: (int           * addrspace(1), int           * addrspace(3), Ii, Ii)
typedef int nu_v2i __attribute__((vector_size(2 * sizeof(int))));
typedef __attribute__((address_space(1))) nu_v2i* nu_g64p;
typedef __attribute__((address_space(3))) nu_v2i* nu_l64p;
typedef __attribute__((address_space(1))) int*    nu_g32p;
typedef __attribute__((address_space(3))) int*    nu_l32p;
#endif

__global__ __launch_bounds__(NU_TPB) void nu_grid_sampler_kernel(
    const float* __restrict__ x,       // [B,C,NX,NY]
    const float* __restrict__ coords,  // [B,N,2]  (coords[...,0]->py axis, [...,1]->px axis)
    const float* __restrict__ s_ptr,   // scalar
    const float* __restrict__ pon,     // [90] pixel_offset_normal (2.5 .. -2.5)
    const float* __restrict__ poi,     // [5]  pixel_offset_indices (-2..2)
    float* __restrict__ out)           // [B,C,N]
{
    __shared__ float s_pon[NU_NRES];
    __shared__ float s_coords[NU_TPB * 2];

    const int t   = threadIdx.x;
    const int tid = blockIdx.x * NU_TPB + t;
    const int b   = tid >> 14;            // / NU_N
    const int n   = tid & (NU_N - 1);     // % NU_N

    // ---- Stage coords tile (2KB) and the 90 offset samples into LDS.
    // Prefer CDNA5 async-to-LDS (ASYNCcnt) when the builtin exists.
#if NU_ASYNC64
    __builtin_amdgcn_global_load_async_to_lds_b64(
        (nu_g64p)(coords + (size_t)tid * 2),
        (nu_l64p)&s_coords[2 * t], 0, 0);
#else
    s_coords[2 * t]     = coords[(size_t)tid * 2];
    s_coords[2 * t + 1] = coords[(size_t)tid * 2 + 1];
#endif
#if NU_ASYNC32
    if (t < NU_NRES) {
        __builtin_amdgcn_global_load_async_to_lds_b32(
            (nu_g32p)(pon + t),
            (nu_l32p)&s_pon[t], 0, 0);
    }
#else
    if (t < NU_NRES) s_pon[t] = pon[t];
#endif

#if NU_ASYNC64 || NU_ASYNC32
    // Issuing wave must drain its own ASYNCcnt before the barrier so other
    // waves see the LDS data after __syncthreads().
#if __has_builtin(__builtin_amdgcn_s_wait_asynccnt)
    __builtin_amdgcn_s_wait_asynccnt(0);
#else
    asm volatile("s_wait_asynccnt 0" ::: "memory");
#endif
#endif
    __syncthreads();

    // ---- Continuous / rounded pixel positions
    const float cy = s_coords[2 * t];       // coords[...,0]
    const float cx = s_coords[2 * t + 1];   // coords[...,1]
    const float px = cx * (float)(NU_NX - 1);
    const float py = cy * (float)(NU_NY - 1);
    const float rpx = rintf(px);            // round-half-even, matches jnp.round
    const float rpy = rintf(py);

    // ---- Integer neighborhood indices (reference clamps BOTH axes with nx-1)
    const int irx = (int)rpx;
    const int iry = (int)rpy;
    int ix[NU_NH], iy[NU_NH];
#pragma unroll
    for (int i = 0; i < NU_NH; ++i) {
        const int o = (int)rintf(poi[i]);   // -2,-1,0,1,2
        ix[i] = min(max(irx - o, 0), NU_NX - 1);
        iy[i] = min(max(iry - o, 0), NU_NX - 1);
    }
    int off[NU_NH * NU_NH];
#pragma unroll
    for (int i = 0; i < NU_NH; ++i)
#pragma unroll
        for (int j = 0; j < NU_NH; ++j)
            off[i * NU_NH + j] = ix[i] * NU_NY + iy[j];

    const float* __restrict__ xb = x + (size_t)b * NU_C * (NU_NX * NU_NY);

    // Prefetch channel-0 neighborhood rows now; the 180-exp weight loop below
    // hides the gather latency. (emits global_prefetch_b8 on gfx1250)
#pragma unroll
    for (int i = 0; i < NU_NH; ++i)
        __builtin_prefetch(xb + off[i * NU_NH + 2], 0, 3);

    // ---- Gaussian bin weights. The 1/(s*sqrt(2pi)) factor cancels exactly in
    // the normalization, so only the exponentials are needed.
    const float s = s_ptr[0];
    const float k = -0.5f / (s * s);
    float wx[NU_NH], wy[NU_NH];
#pragma unroll
    for (int i = 0; i < NU_NH; ++i) {
        float ax = 0.0f, ay = 0.0f;
        for (int r = 0; r < NU_BIN; ++r) {
            const float o  = s_pon[i * NU_BIN + r];
            // reference: clip(rp - offset, 0, nx) on BOTH axes (nx quirk kept)
            const float vx = fminf(fmaxf(rpx - o, 0.0f), (float)NU_NX);
            const float vy = fminf(fmaxf(rpy - o, 0.0f), (float)NU_NX);
            const float dx = vx - px;
            const float dy = vy - py;
            ax += __expf(k * dx * dx);
            ay += __expf(k * dy * dy);
        }
        wx[i] = ax;
        wy[i] = ay;
    }
    const float sx  = wx[0] + wx[1] + wx[2] + wx[3] + wx[4];
    const float sy  = wy[0] + wy[1] + wy[2] + wy[3] + wy[4];
    const float inv = 1.0f / (sx * sy);
#pragma unroll
    for (int i = 0; i < NU_NH; ++i) wx[i] *= inv;   // fold normalization into wx

    // ---- Gather + separable weighted reduction, all 16 channels
    const size_t outbase = ((size_t)b * NU_C) * NU_N + (size_t)n;
    for (int c = 0; c < NU_C; ++c) {
        const float* __restrict__ xc = xb + (size_t)c * (NU_NX * NU_NY);
        float acc = 0.0f;
#pragma unroll
        for (int i = 0; i < NU_NH; ++i) {
            float ti = 0.0f;
#pragma unroll
            for (int j = 0; j < NU_NH; ++j)
                ti = fmaf(xc[off[i * NU_NH + j]], wy[j], ti);
            acc = fmaf(ti, wx[i], acc);
        }
        // streamed output: non-temporal so L2 stays dedicated to x
        __builtin_nontemporal_store(acc, &out[outbase + (size_t)c * NU_N]);
    }
}

extern "C" void kernel_launch(void* const* d_in, const int* in_sizes, int n_in,
                              void* d_out, int out_size, void* d_ws, size_t ws_size,
                              hipStream_t stream) {
    const float* x      = (const float*)d_in[0];
    const float* coords = (const float*)d_in[1];
    const float* s      = (const float*)d_in[2];
    const float* pon    = (const float*)d_in[3];
    const float* poi    = (const float*)d_in[4];
    float* out          = (float*)d_out;

    const int total  = NU_B * NU_N;            // 131072 threads, one per (b,n)
    const int blocks = total / NU_TPB;         // 512 blocks of 256 (8 wave32)
    nu_grid_sampler_kernel<<<blocks, NU_TPB, 0, stream>>>(x, coords, s, pon, poi, out);
}